// LightPrompt_34428457844927
// MI455X (gfx1250) — compile-verified
//
#include <hip/hip_runtime.h>

typedef __attribute__((ext_vector_type(2))) float v2f;
typedef __attribute__((ext_vector_type(4))) float v4f;
typedef __attribute__((ext_vector_type(8))) float v8f;

#define TT 4096       // tokens per group
#define DD 128        // feature dim
#define EE 4          // edge_token dim
#define BM 128        // block tile rows
#define BN 128        // block tile cols
#define LDS_STRIDE 132  // 128 + 4 pad: row stride % 64 banks == 4 -> conflict-free frag loads
#define THRE 0.55f
#define NEG_SLOPE 0.01f

__device__ __forceinline__ float leaky(float v) {
    return v >= 0.0f ? v : NEG_SLOPE * v;
}

__global__ __launch_bounds__(256) void lightprompt_gram_kernel(
    const float* __restrict__ tokens,      // [TT, DD]
    const float* __restrict__ edge_token,  // [EE]
    float* __restrict__ out_x,             // [TT*DD]
    float* __restrict__ out_adj,           // [TT*TT]
    float* __restrict__ out_attr)          // [TT*TT*EE]
{
    extern __shared__ float lds[];
    float* As = lds;                       // [BM][LDS_STRIDE]
    float* Bs = lds + BM * LDS_STRIDE;     // [BN][LDS_STRIDE]

    const int tid    = threadIdx.x;
    const int wid    = tid >> 5;           // wave32
    const int lane   = tid & 31;
    const int laneLo = lane & 15;
    const int laneHi = lane >> 4;

    const int rowBase = blockIdx.y * BM;
    const int colBase = blockIdx.x * BN;

    // ---- fold in the x = tokens identity copy (0.5M floats over 1024 blocks) ----
    {
        int bid = blockIdx.y * gridDim.x + blockIdx.x;   // 0..1023
        if (tid < 128) {
            int i4 = bid * 128 + tid;                    // 1024*128 = 131072 float4 = all of x
            v4f v = *(const v4f*)(tokens + i4 * 4);
            *(v4f*)(out_x + i4 * 4) = v;
        }
    }

    // ---- stage A (row block) and B (col block) tiles into padded LDS ----
    // 128 rows x 32 float4 per tile = 4096 float4; 256 threads -> 16 each
    #pragma unroll 4
    for (int i = tid; i < BM * (DD / 4); i += 256) {
        int r  = i >> 5;       // /32
        int c4 = i & 31;
        v4f v = *(const v4f*)(tokens + (size_t)(rowBase + r) * DD + c4 * 4);
        *(v4f*)(As + r * LDS_STRIDE + c4 * 4) = v;
    }
    #pragma unroll 4
    for (int i = tid; i < BN * (DD / 4); i += 256) {
        int r  = i >> 5;
        int c4 = i & 31;
        v4f v = *(const v4f*)(tokens + (size_t)(colBase + r) * DD + c4 * 4);
        *(v4f*)(Bs + r * LDS_STRIDE + c4 * 4) = v;
    }
    __syncthreads();

    // ---- per-wave 64x32 region: 4 row-subtiles x 2 col-subtiles of 16x16 ----
    const int waveRow = (wid & 1) * 64;
    const int waveCol = (wid >> 1) * 32;

    v8f acc[4][2];
    #pragma unroll
    for (int i = 0; i < 4; ++i)
        #pragma unroll
        for (int j = 0; j < 2; ++j)
            acc[i][j] = (v8f){0.f, 0.f, 0.f, 0.f, 0.f, 0.f, 0.f, 0.f};

    // K loop: 32 WMMA steps of K=4.
    // fp32 16x4 A layout: lanes 0-15 -> K={k0,k0+1}, lanes 16-31 -> K={k0+2,k0+3}
    #pragma unroll 4
    for (int k0 = 0; k0 < DD; k0 += 4) {
        const int kk = k0 + 2 * laneHi;
        v2f a[4], b[2];
        #pragma unroll
        for (int i = 0; i < 4; ++i)
            a[i] = *(const v2f*)(As + (waveRow + i * 16 + laneLo) * LDS_STRIDE + kk);
        #pragma unroll
        for (int j = 0; j < 2; ++j)
            b[j] = *(const v2f*)(Bs + (waveCol + j * 16 + laneLo) * LDS_STRIDE + kk);

        #pragma unroll
        for (int i = 0; i < 4; ++i)
            #pragma unroll
            for (int j = 0; j < 2; ++j)
                acc[i][j] = __builtin_amdgcn_wmma_f32_16x16x4_f32(
                    /*neg_a=*/false, a[i], /*neg_b=*/false, b[j],
                    /*c_mod=*/(short)0, acc[i][j],
                    /*reuse_a=*/false, /*reuse_b=*/false);
    }

    // ---- epilogue: sigmoid, prune, leaky_relu edge attrs ----
    const float et0 = edge_token[0];
    const float et1 = edge_token[1];
    const float et2 = edge_token[2];
    const float et3 = edge_token[3];

    #pragma unroll
    for (int i = 0; i < 4; ++i) {
        #pragma unroll
        for (int j = 0; j < 2; ++j) {
            #pragma unroll
            for (int r = 0; r < 8; ++r) {
                // C layout: VGPR r, lanes 0-15 -> M=r, lanes 16-31 -> M=r+8; N = laneLo
                const int row = rowBase + waveRow + i * 16 + r + 8 * laneHi;
                const int col = colBase + waveCol + j * 16 + laneLo;

                const float dot  = acc[i][j][r];
                const float sim  = 1.0f / (1.0f + __expf(-dot));
                const bool  edge = sim >= THRE;
                const float m    = edge ? 1.0f : 0.0f;

                out_adj[row * TT + col] = edge ? sim : 0.0f;

                v4f attr;
                attr.x = leaky(dot * et0) * m;
                attr.y = leaky(dot * et1) * m;
                attr.z = leaky(dot * et2) * m;
                attr.w = leaky(dot * et3) * m;
                *(v4f*)(out_attr + ((size_t)(row * TT + col)) * EE) = attr;
            }
        }
    }
}

extern "C" void kernel_launch(void* const* d_in, const int* in_sizes, int n_in,
                              void* d_out, int out_size, void* d_ws, size_t ws_size,
                              hipStream_t stream) {
    (void)in_sizes; (void)n_in; (void)out_size; (void)d_ws; (void)ws_size;

    const float* tokens     = (const float*)d_in[0];  // [1, 4096, 128]
    const float* edge_token = (const float*)d_in[1];  // [1, 4]

    float* out_x    = (float*)d_out;                              // 4096*128
    float* out_adj  = out_x + (size_t)TT * DD;                    // 4096*4096
    float* out_attr = out_adj + (size_t)TT * TT;                  // 4096*4096*4

    dim3 grid(TT / BN, TT / BM);   // 32 x 32
    dim3 block(256);
    size_t smem = (size_t)(BM + BN) * LDS_STRIDE * sizeof(float); // 132 KB dynamic LDS

    lightprompt_gram_kernel<<<grid, block, smem, stream>>>(
        tokens, edge_token, out_x, out_adj, out_attr);
}